// SpikeMixtureModel_59760174957355
// MI455X (gfx1250) — compile-verified
//
#include <hip/hip_runtime.h>
#include <hip/hip_bf16.h>

// ---------------------------------------------------------------------------
// SpikeMixtureModel for MI455X (gfx1250, wave32).
//
// emp_mean[r,c] = (sum over (n,a) with chan[n,a]==c, f[n,0,a] finite of
//                  w[n]*f[n,r,a]) / (Wtot - sum over NaN-hit weights at c)
// assignments[j] = min row among max of (lik_data - (min(lik_data)-1)) in
//                  column j (10 contiguous entries per column).
//
// ~50MB traffic -> ~2.2us at 23.3 TB/s; bandwidth/latency bound, f32 only.
// Determinism: per-wave private LDS accumulators (channels within a spike are
// distinct -> no intra-wave conflicts), fixed-order tree reductions, no float
// atomics anywhere.
// CDNA5 paths: global_load_async_to_lds_b128/b32 4-deep pipelined feature
// staging (ASYNCcnt-tracked) + V_WMMA_F32_16X16X4_F32 cross-block reduction.
// ---------------------------------------------------------------------------

#define RANKS   5
#define NA      32
#define NCH     384
#define ACC     (RANKS * NCH + NCH)   // 2304 floats per accumulator copy
#define PARTPAD 2320                  // ACC + wsum + zero pad, multiple of 16
#define NBLK1   512                   // phase-1 blocks (2048 waves)
#define THREADS1 128                  // 4 waves per block
#define NWAVE   4
#define NBUF    4                     // async pipeline depth (3 ahead)
#define NMINB   256
#define NNZ_PER 10

// ---- CDNA5 async-to-LDS path (probe via __has_builtin) --------------------
#if __has_builtin(__builtin_amdgcn_global_load_async_to_lds_b32)
#define HAVE_ASYNC 1
#else
#define HAVE_ASYNC 0
#endif
#if __has_builtin(__builtin_amdgcn_global_load_async_to_lds_b128)
#define HAVE_ASYNC128 1
#define ASYNC_P 2                     // async instructions per spike prefetch
#else
#define HAVE_ASYNC128 0
#define ASYNC_P 5
#endif
#if __has_builtin(__builtin_amdgcn_s_wait_asynccnt)
#define WAIT_ASYNC(n) __builtin_amdgcn_s_wait_asynccnt(n)
#else
#define WAIT_ASYNC(n) asm volatile("s_wait_asynccnt %0" ::"i"(n) : "memory")
#endif

typedef float v2f __attribute__((ext_vector_type(2)));
typedef float v8f __attribute__((ext_vector_type(8)));
typedef int   i32x4 __attribute__((vector_size(16)));   // clang builtin V4i

// ---------------------------------------------------------------------------
// Phase 1: per-block partial accumulation.
// Wave w owns LDS copy sm[w*ACC .. w*ACC+ACC). One spike per wave-iteration:
// lane a handles channel slot a (distinct channels -> conflict-free LDS RMW).
// Features staged through LDS with a 4-buffer async pipeline, 3 spikes ahead.
// ---------------------------------------------------------------------------
__global__ __launch_bounds__(THREADS1)
void smm_accum(const float* __restrict__ feats, const int* __restrict__ chans,
               const float* __restrict__ wts, float* __restrict__ part, int n) {
  __shared__ float sm[NWAVE * ACC];
  __shared__ float swsum[NWAVE];
#if HAVE_ASYNC
  __shared__ alignas(16) float stage[NWAVE][NBUF][RANKS * NA];
#endif
  const int tid  = threadIdx.x;
  const int wave = tid >> 5;
  const int lane = tid & 31;

  for (int i = tid; i < NWAVE * ACC; i += THREADS1) sm[i] = 0.0f;
  if (tid < NWAVE) swsum[tid] = 0.0f;
  __syncthreads();

  float* acc = sm + wave * ACC;
  const int spb = (n + NBLK1 - 1) / NBLK1;    // spikes per block
  const int spw = (spb + NWAVE - 1) / NWAVE;  // spikes per wave
  int blk_end = blockIdx.x * spb + spb; if (blk_end > n) blk_end = n;
  int s0 = blockIdx.x * spb + wave * spw;
  int s1 = s0 + spw; if (s1 > blk_end) s1 = blk_end;

  float wsum = 0.0f;

#if HAVE_ASYNC
  typedef __attribute__((address_space(1))) int   GI;   // __device__ int*
  typedef __attribute__((address_space(3))) int   LI;   // __shared__ int*
  typedef __attribute__((address_space(1))) i32x4 GV4;  // __device__ v4i*
  typedef __attribute__((address_space(3))) i32x4 LV4;  // __shared__ v4i*
  auto prefetch = [&](int s, int buf) {
    const float* g = feats + (size_t)s * (RANKS * NA);
    float* l = &stage[wave][buf][0];
#if HAVE_ASYNC128
    __builtin_amdgcn_global_load_async_to_lds_b128((GV4*)(g + 4 * lane),
                                                   (LV4*)(l + 4 * lane), 0, 0);
    __builtin_amdgcn_global_load_async_to_lds_b32((GI*)(g + 128 + lane),
                                                  (LI*)(l + 128 + lane), 0, 0);
#else
#pragma unroll
    for (int r = 0; r < RANKS; ++r)
      __builtin_amdgcn_global_load_async_to_lds_b32(
          (GI*)(g + r * NA + lane), (LI*)(l + r * NA + lane), 0, 0);
#endif
  };
#pragma unroll
  for (int p = 0; p < NBUF - 1; ++p)
    if (s0 + p < s1) prefetch(s0 + p, p);
#endif

  for (int s = s0; s < s1; ++s) {
    float wt = wts[s];
    int c = chans[s * NA + lane];
    float fv[RANKS];
#if HAVE_ASYNC
    const int cb = (s - s0) & (NBUF - 1);
    if (s + (NBUF - 1) < s1) prefetch(s + (NBUF - 1), (cb + NBUF - 1) & (NBUF - 1));
    // Wait until this spike's loads are done: allow only the newer prefetches
    // (exact taper in the tail so we never skip a wait on our own loads).
    const int rem = s1 - 1 - s;
    if (rem >= 3)      WAIT_ASYNC(3 * ASYNC_P);
    else if (rem == 2) WAIT_ASYNC(2 * ASYNC_P);
    else if (rem == 1) WAIT_ASYNC(1 * ASYNC_P);
    else               WAIT_ASYNC(0);
    asm volatile("" ::: "memory");
    const float* cur = &stage[wave][cb][0];
#pragma unroll
    for (int r = 0; r < RANKS; ++r) fv[r] = cur[r * NA + lane];
#else
    const float* f = feats + (size_t)s * (RANKS * NA);
#pragma unroll
    for (int r = 0; r < RANKS; ++r) fv[r] = f[r * NA + lane];
#endif
    if (lane == 0) wsum += wt;
    if (__builtin_isfinite(fv[0])) {
#pragma unroll
      for (int r = 0; r < RANKS; ++r) acc[r * NCH + c] += wt * fv[r];
    } else {
      acc[RANKS * NCH + c] += wt;   // NaN-hit weight per channel
    }
  }
  if (lane == 0) swsum[wave] = wsum;
  __syncthreads();

  // Deterministic cross-wave reduce; write padded block partial record.
  float* dst = part + (size_t)blockIdx.x * PARTPAD;
  for (int i = tid; i < ACC; i += THREADS1) {
    float t = 0.0f;
#pragma unroll
    for (int w = 0; w < NWAVE; ++w) t += sm[w * ACC + i];
    dst[i] = t;
  }
  if (tid == 0) {
    float t = 0.0f;
#pragma unroll
    for (int w = 0; w < NWAVE; ++w) t += swsum[w];
    dst[ACC] = t;
  }
  for (int i = ACC + 1 + tid; i < PARTPAD; i += THREADS1) dst[i] = 0.0f;
}

// ---------------------------------------------------------------------------
// Phase 2: cross-block reduction via V_WMMA_F32_16X16X4_F32.
// D = A(16 elems x 4 block-partials) * B(ones) + C  => every column of D is
// the row sum; accumulate over nblk/4 steps. A layout per ISA: lanes 0-15 hold
// K=0,1 in VGPR0/1; lanes 16-31 hold K=2,3. B=ones makes its layout moot.
// PARTPAD is a multiple of 16 and zero-padded, so no bounds checks anywhere.
// ---------------------------------------------------------------------------
__global__ __launch_bounds__(32)
void smm_reduce_wmma(const float* __restrict__ part, float* __restrict__ tot,
                     int nblk) {
  const int lane = threadIdx.x;
  const int li = lane & 15;
  const int hi = lane >> 4;
  const int e = blockIdx.x * 16 + li;     // element this lane loads (row M=li)
  v8f c = {};
  v2f bv; bv.x = 1.0f; bv.y = 1.0f;
  for (int k0 = 0; k0 < nblk; k0 += 4) {
    const int kk = k0 + 2 * hi;
    v2f a;
    a.x = part[(size_t)kk * PARTPAD + e];
    a.y = part[(size_t)(kk + 1) * PARTPAD + e];
    c = __builtin_amdgcn_wmma_f32_16x16x4_f32(false, a, false, bv, (short)0, c,
                                              false, false);
  }
  // D layout: c[v] at lane L = D[v + 8*(L>>4)][L&15]; columns are identical.
  if (li < 8) {
    float v;
    switch (li) {
      case 0: v = c[0]; break; case 1: v = c[1]; break;
      case 2: v = c[2]; break; case 3: v = c[3]; break;
      case 4: v = c[4]; break; case 5: v = c[5]; break;
      case 6: v = c[6]; break; default: v = c[7]; break;
    }
    tot[blockIdx.x * 16 + hi * 8 + li] = v;   // M = li + 8*hi
  }
}

// Finalize emp_mean = num / (Wtot - nanw[c]).
__global__ void smm_final(const float* __restrict__ tot, float* __restrict__ out) {
  int idx = blockIdx.x * blockDim.x + threadIdx.x;
  if (idx < RANKS * NCH) {
    float num  = tot[idx];
    float nanw = tot[RANKS * NCH + (idx % NCH)];
    float wtot = tot[ACC];
    out[idx] = num / (wtot - nanw);
  }
}

// ---------------------------------------------------------------------------
// Assignment phase: global min of lik_data (exact, order-independent), then
// per-column argmax-with-min-row-tiebreak in the shifted domain.
// ---------------------------------------------------------------------------
__global__ __launch_bounds__(256)
void smm_min1(const float* __restrict__ data, float* __restrict__ part, int n) {
  __shared__ float sm[256];
  float m = __builtin_inff();
  for (int i = blockIdx.x * 256 + threadIdx.x; i < n; i += gridDim.x * 256)
    m = fminf(m, data[i]);
  sm[threadIdx.x] = m;
  __syncthreads();
  for (int s = 128; s > 0; s >>= 1) {
    if (threadIdx.x < s) sm[threadIdx.x] = fminf(sm[threadIdx.x], sm[threadIdx.x + s]);
    __syncthreads();
  }
  if (threadIdx.x == 0) part[blockIdx.x] = sm[0];
}

__global__ __launch_bounds__(256)
void smm_min2(const float* __restrict__ part, float* __restrict__ offp, int nb) {
  __shared__ float sm[256];
  float m = __builtin_inff();
  for (int i = threadIdx.x; i < nb; i += 256) m = fminf(m, part[i]);
  sm[threadIdx.x] = m;
  __syncthreads();
  for (int s = 128; s > 0; s >>= 1) {
    if (threadIdx.x < s) sm[threadIdx.x] = fminf(sm[threadIdx.x], sm[threadIdx.x + s]);
    __syncthreads();
  }
  if (threadIdx.x == 0) offp[0] = sm[0] - 1.0f;
}

__global__ __launch_bounds__(256)
void smm_assign(const float* __restrict__ data, const int* __restrict__ rows,
                const float* __restrict__ offp, float* __restrict__ out, int n) {
  int j = blockIdx.x * 256 + threadIdx.x;
  if (j >= n) return;
  const float off = offp[0];
  float smax = -__builtin_inff();
  int best = 0x7fffffff;
#pragma unroll
  for (int i = 0; i < NNZ_PER; ++i) {
    float s = data[j * NNZ_PER + i] - off;  // compare in shifted domain
    int r = rows[j * NNZ_PER + i];
    if (s > smax)       { smax = s; best = r; }
    else if (s == smax) { best = (r < best) ? r : best; }
  }
  out[j] = (best == 0x7fffffff) ? -1.0f : (float)best;
}

// ---------------------------------------------------------------------------
extern "C" void kernel_launch(void* const* d_in, const int* in_sizes, int n_in,
                              void* d_out, int out_size, void* d_ws, size_t ws_size,
                              hipStream_t stream) {
  const float* feats    = (const float*)d_in[0];  // (n,5,32)
  const int*   chans    = (const int*)  d_in[1];  // (n,32)
  const float* wts      = (const float*)d_in[2];  // (n,)
  const int*   lik_rows = (const int*)  d_in[3];  // (nnz,)
  // d_in[4] = lik_cols: structurally repeat(arange(n), 10) -> contiguous groups
  const float* lik_data = (const float*)d_in[5];  // (nnz,)
  const int n   = in_sizes[2];
  const int nnz = in_sizes[5];

  float* out   = (float*)d_out;             // [0,1920): emp_mean; [1920,+n): assignments
  float* wsf   = (float*)d_ws;
  float* part  = wsf;                       // NBLK1 * PARTPAD floats (~4.8 MB)
  float* tot   = part + (size_t)NBLK1 * PARTPAD;
  float* minp  = tot + PARTPAD;             // NMINB floats
  float* offp  = minp + NMINB;              // 1 float
  (void)n_in; (void)out_size; (void)ws_size;

  smm_accum<<<NBLK1, THREADS1, 0, stream>>>(feats, chans, wts, part, n);
  smm_reduce_wmma<<<PARTPAD / 16, 32, 0, stream>>>(part, tot, NBLK1);
  smm_final<<<(RANKS * NCH + 255) / 256, 256, 0, stream>>>(tot, out);
  smm_min1<<<NMINB, 256, 0, stream>>>(lik_data, minp, nnz);
  smm_min2<<<1, 256, 0, stream>>>(minp, offp, NMINB);
  smm_assign<<<(n + 255) / 256, 256, 0, stream>>>(lik_data, lik_rows, offp,
                                                  out + RANKS * NCH, n);
}